// PriorBoxGenerator3D_15006615733620
// MI455X (gfx1250) — compile-verified
//
#include <hip/hip_runtime.h>
#include <stdint.h>

// Problem constants (from reference): scores (8,1,1024,1024) f32, boxes (8,7,1024,1024) f32
#define NBATCH 8
#define HW     (1u << 20)   // 1024*1024
#define CBOX   7
#define TOPK   100
#define NB     4096          // histogram buckets = top 12 bits of monotone key
#define CAP    4096          // candidate capacity per batch

// ws layout (uint32 units unless noted)
#define HIST_U32 (NBATCH * NB)                  // 32768 uints
#define THR_OFF  (HIST_U32)                     // NBATCH uints
#define CNT_OFF  (THR_OFF + NBATCH)             // NBATCH uints
#define KEY_BYTE_OFF ((size_t)(CNT_OFF + NBATCH) * 4)  // 131136, 8-byte aligned

#if defined(__HIP_DEVICE_COMPILE__) && defined(__gfx1250__) && \
    __has_builtin(__builtin_amdgcn_global_load_async_to_lds_b128) && \
    __has_builtin(__builtin_amdgcn_s_wait_asynccnt)
#define USE_ASYNC_LDS 1
#else
#define USE_ASYNC_LDS 0
#endif

// Compile-time probe: require the CDNA5 async-to-LDS path on the gfx1250
// device pass. If the builtins are missing this errors instead of silently
// falling back, so a clean compile == async instructions are in the binary.
#if defined(__HIP_DEVICE_COMPILE__) && defined(__gfx1250__) && !USE_ASYNC_LDS
#error "gfx1250 async global->LDS builtins not available on this toolchain"
#endif

typedef int v4i __attribute__((ext_vector_type(4)));

__device__ __forceinline__ unsigned mono_key(float f) {
    unsigned b = __float_as_uint(f);
    return (b & 0x80000000u) ? ~b : (b | 0x80000000u);
}
__device__ __forceinline__ float unmono_key(unsigned u) {
    unsigned b = (u & 0x80000000u) ? (u ^ 0x80000000u) : ~u;
    return __uint_as_float(b);
}

// ---------------- Kernel 0: zero histogram + counters -----------------------
__global__ void rpn_init_kernel(unsigned* __restrict__ ws) {
    int i = blockIdx.x * blockDim.x + threadIdx.x;
    int total = HIST_U32 + 2 * NBATCH;   // hist + thr + cnt
    if (i < total) ws[i] = 0u;
}

// ---------------- Kernel 1: 12-bit histogram of monotone keys ---------------
// grid = (128, NBATCH), block = 256. 1024 tiles/batch, 8 tiles/block.
// Streams scores through LDS with double-buffered async global->LDS loads.
__global__ void rpn_hist_kernel(const float* __restrict__ scores,
                                unsigned* __restrict__ hist) {
    const int b   = blockIdx.y;
    const int tid = threadIdx.x;
    __shared__ unsigned lhist[NB];
    for (int i = tid; i < NB; i += blockDim.x) lhist[i] = 0u;
    __syncthreads();

    const float* base = scores + (size_t)b * HW;
    const int tiles = HW / (256 * 4);    // 1024 tiles of 256 lanes x float4

#if USE_ASYNC_LDS
    __shared__ float stage[2][256 * 4];  // 2 x 4KB staging buffers
    auto issue = [&](int tile, int bufi) {
        const float* gp = base + ((size_t)tile * 256 + tid) * 4;
        float* lp = &stage[bufi][tid * 4];
        __builtin_amdgcn_global_load_async_to_lds_b128(
            (v4i __attribute__((address_space(1)))*)gp,
            (v4i __attribute__((address_space(3)))*)lp,
            0, 0);
    };
    int t = blockIdx.x, step = gridDim.x, buf = 0;
    if (t < tiles) issue(t, buf);
    for (; t < tiles; t += step) {
        int nxt = t + step;
        if (nxt < tiles) {
            issue(nxt, buf ^ 1);
            __builtin_amdgcn_s_wait_asynccnt(1);   // oldest (current tile) done
        } else {
            __builtin_amdgcn_s_wait_asynccnt(0);
        }
        float4 v = *(const float4*)&stage[buf][tid * 4];
        atomicAdd(&lhist[mono_key(v.x) >> 20], 1u);
        atomicAdd(&lhist[mono_key(v.y) >> 20], 1u);
        atomicAdd(&lhist[mono_key(v.z) >> 20], 1u);
        atomicAdd(&lhist[mono_key(v.w) >> 20], 1u);
        buf ^= 1;
    }
#else
    const float4* b4 = (const float4*)base;
    for (int i = blockIdx.x * blockDim.x + tid; i < (int)(HW / 4);
         i += gridDim.x * blockDim.x) {
        float4 v = b4[i];
        atomicAdd(&lhist[mono_key(v.x) >> 20], 1u);
        atomicAdd(&lhist[mono_key(v.y) >> 20], 1u);
        atomicAdd(&lhist[mono_key(v.z) >> 20], 1u);
        atomicAdd(&lhist[mono_key(v.w) >> 20], 1u);
    }
    (void)tiles;
#endif
    __syncthreads();
    for (int i = tid; i < NB; i += blockDim.x) {
        unsigned v = lhist[i];
        if (v) atomicAdd(&hist[(size_t)b * NB + i], v);
    }
}

// ---------------- Kernel 2: suffix-scan histogram -> threshold bucket -------
__global__ void rpn_thresh_kernel(const unsigned* __restrict__ hist,
                                  unsigned* __restrict__ thr) {
    int b = blockIdx.x;
    if (threadIdx.x == 0) {
        unsigned cum = 0;
        int t = NB - 1;
        for (; t >= 0; --t) {
            cum += hist[(size_t)b * NB + t];
            if (cum >= TOPK) break;
        }
        if (t < 0) t = 0;
        thr[b] = (unsigned)t;
    }
}

// ---------------- Kernel 3: compact candidates (L2-hot re-scan) -------------
__global__ void rpn_collect_kernel(const float* __restrict__ scores,
                                   const unsigned* __restrict__ thr,
                                   unsigned* __restrict__ cnt,
                                   unsigned long long* __restrict__ keys) {
    const int b = blockIdx.y;
    const unsigned T = thr[b];
    const float4* base = (const float4*)(scores + (size_t)b * HW);
    for (int i = blockIdx.x * blockDim.x + threadIdx.x; i < (int)(HW / 4);
         i += gridDim.x * blockDim.x) {
        float4 v = base[i];
        unsigned p0 = 4u * (unsigned)i;
        float f[4] = {v.x, v.y, v.z, v.w};
        #pragma unroll
        for (int k = 0; k < 4; ++k) {
            unsigned u = mono_key(f[k]);
            if ((u >> 20) >= T) {
                unsigned pos = atomicAdd(&cnt[b], 1u);
                if (pos < CAP) {
                    // high 32 = monotone score key; low 32 = ~index
                    // => larger key == larger score, ties -> smaller index
                    keys[(size_t)b * CAP + pos] =
                        ((unsigned long long)u << 32) | (unsigned)(~(p0 + k));
                }
            }
        }
    }
}

// ---------------- Kernel 4: per-batch top-100 selection + gather ------------
__global__ void rpn_topk_kernel(const float* __restrict__ boxes,
                                const unsigned* __restrict__ cnt,
                                const unsigned long long* __restrict__ keys,
                                float* __restrict__ out) {
    const int b = blockIdx.x;
    const int tid = threadIdx.x;
    __shared__ unsigned long long lk[CAP];
    __shared__ unsigned long long red[256];
    __shared__ unsigned selIdx[TOPK];
    __shared__ unsigned selU[TOPK];

    unsigned n = cnt[b];
    if (n > CAP) n = CAP;
    for (unsigned i = tid; i < n; i += 256) lk[i] = keys[(size_t)b * CAP + i];
    __syncthreads();

    for (int j = 0; j < TOPK; ++j) {
        unsigned long long m = 0ull;
        for (unsigned i = tid; i < n; i += 256) {
            unsigned long long v = lk[i];
            if (v > m) m = v;
        }
        red[tid] = m;
        __syncthreads();
        for (int s = 128; s > 0; s >>= 1) {
            if (tid < s) {
                unsigned long long o = red[tid + s];
                if (o > red[tid]) red[tid] = o;
            }
            __syncthreads();
        }
        unsigned long long best = red[0];
        if (tid == 0) {
            selU[j]   = (unsigned)(best >> 32);
            unsigned idx = ~(unsigned)best;
            selIdx[j] = (idx < HW) ? idx : 0u;  // safety clamp
        }
        // remove winner (keys are unique since index is unique)
        for (unsigned i = tid; i < n; i += 256)
            if (lk[i] == best) lk[i] = 0ull;
        __syncthreads();
    }

    // gather: 100 proposals x (7 box channels + 1 score) = 800 writes
    for (int w = tid; w < TOPK * 8; w += 256) {
        int j = w >> 3, c = w & 7;
        unsigned idx = selIdx[j];
        if (c < 7) {
            // proposals[b][j][c] = boxes[b][c][idx]
            out[((size_t)b * TOPK + j) * CBOX + c] =
                boxes[((size_t)(b * CBOX + c)) * HW + idx];
        } else {
            // scores after all proposals: offset = NBATCH*TOPK*CBOX
            out[(size_t)NBATCH * TOPK * CBOX + (size_t)b * TOPK + j] =
                unmono_key(selU[j]);
        }
    }
}

extern "C" void kernel_launch(void* const* d_in, const int* in_sizes, int n_in,
                              void* d_out, int out_size, void* d_ws, size_t ws_size,
                              hipStream_t stream) {
    const float* scores = (const float*)d_in[0];
    const float* boxes  = (const float*)d_in[1];
    float* out = (float*)d_out;

    unsigned* ws   = (unsigned*)d_ws;
    unsigned* hist = ws;
    unsigned* thr  = ws + THR_OFF;
    unsigned* cnt  = ws + CNT_OFF;
    unsigned long long* keys =
        (unsigned long long*)((char*)d_ws + KEY_BYTE_OFF);

    int initN = HIST_U32 + 2 * NBATCH;
    rpn_init_kernel<<<(initN + 255) / 256, 256, 0, stream>>>(ws);
    rpn_hist_kernel<<<dim3(128, NBATCH), 256, 0, stream>>>(scores, hist);
    rpn_thresh_kernel<<<NBATCH, 32, 0, stream>>>(hist, thr);
    rpn_collect_kernel<<<dim3(128, NBATCH), 256, 0, stream>>>(scores, thr, cnt, keys);
    rpn_topk_kernel<<<NBATCH, 256, 0, stream>>>(boxes, cnt, keys, out);
}